// InteractionBlock_4647154614870
// MI455X (gfx1250) — compile-verified
//
#include <hip/hip_runtime.h>

// MI455X / gfx1250, wave32. All matrix math via V_WMMA_F32_16X16X4_F32 (exact fp32).

typedef float v2f __attribute__((ext_vector_type(2)));
typedef float v8f __attribute__((ext_vector_type(8)));

#define HIDN 128
#define NGC  50
#define LOG2F_ 0.69314718055994530942f

// Non-returning fp32 global atomic add: relaxed, device scope -> GLOBAL_ATOMIC_ADD_F32
__device__ __forceinline__ void atomic_add_f32(float* p, float v) {
  (void)__hip_atomic_fetch_add(p, v, __ATOMIC_RELAXED, __HIP_MEMORY_SCOPE_AGENT);
}

__device__ __forceinline__ v8f wmma4(v2f a, v2f b, v8f c) {
  // (neg_a, A, neg_b, B, c_mod, C, reuse_a, reuse_b)
  return __builtin_amdgcn_wmma_f32_16x16x4_f32(false, a, false, b, (short)0, c,
                                               false, false);
}

__device__ __forceinline__ float ssp(float v) {
  // shifted softplus: log(1+exp(v)) - log(2), numerically stable
  return fmaxf(v, 0.0f) + log1pf(__expf(-fabsf(v))) - LOG2F_;
}

__device__ __forceinline__ void zero_acc(v8f acc[8]) {
#pragma unroll
  for (int jt = 0; jt < 8; ++jt)
    acc[jt] = (v8f){0.f, 0.f, 0.f, 0.f, 0.f, 0.f, 0.f, 0.f};
}

// ---------------------------------------------------------------------------
// aggr := 0
__global__ void zero_kernel(float* __restrict__ p, long n) {
  long i = (long)blockIdx.x * blockDim.x + threadIdx.x;
  long stride = (long)gridDim.x * blockDim.x;
  for (; i < n; i += stride) p[i] = 0.0f;
}

// ---------------------------------------------------------------------------
// xf = x @ lin1_w.T   (M=N rows, K=128, N=128). One wave -> 16 rows x 128 cols.
__global__ __launch_bounds__(256) void xf_kernel(const float* __restrict__ x,
                                                 const float* __restrict__ w,
                                                 float* __restrict__ xf,
                                                 int nrows) {
  const int lane = threadIdx.x & 31;
  const int m = lane & 15;
  const int half = lane >> 4;
  const int koff = half * 2;
  const int wid = blockIdx.x * (blockDim.x >> 5) + (threadIdx.x >> 5);
  const int rbase = wid * 16;
  if (rbase >= nrows) return;

  int arow = rbase + m;
  if (arow > nrows - 1) arow = nrows - 1;
  const float* A = x + (size_t)arow * HIDN;

  v8f acc[8];
  zero_acc(acc);

  for (int kk = 0; kk < HIDN; kk += 4) {
    v2f a = *(const v2f*)(A + kk + koff);
#pragma unroll
    for (int jt = 0; jt < 8; ++jt) {
      v2f b = *(const v2f*)(w + (size_t)(jt * 16 + m) * HIDN + kk + koff);
      acc[jt] = wmma4(a, b, acc[jt]);
    }
  }

#pragma unroll
  for (int g = 0; g < 8; ++g) {
    int row = rbase + g + 8 * half;
    if (row < nrows) {
#pragma unroll
      for (int jt = 0; jt < 8; ++jt)
        xf[(size_t)row * HIDN + jt * 16 + m] = acc[jt][g];
    }
  }
}

// ---------------------------------------------------------------------------
// Fused edge pipeline: h = ssp(edge_attr @ w1.T + b1); W = h @ w2.T + b2;
// then scatter: aggr[col] += W * xf[row]; aggr[row] += W * xf[col].
// W stays in registers (never written to memory). 4 waves / block.
__global__ __launch_bounds__(128) void edge_kernel(
    const float* __restrict__ eattr, const int* __restrict__ ei,
    const float* __restrict__ w1, const float* __restrict__ b1,
    const float* __restrict__ w2, const float* __restrict__ b2,
    const float* __restrict__ xf, float* __restrict__ aggr, int nedges) {
  __shared__ float hbuf[4 * 16 * 132];  // per-wave 16x128 tile, row stride 132

  const int lane = threadIdx.x & 31;
  const int m = lane & 15;
  const int half = lane >> 4;
  const int koff = half * 2;
  const int wslot = threadIdx.x >> 5;
  const int wid = blockIdx.x * 4 + wslot;
  const int ebase = wid * 16;
  if (ebase >= nedges) return;
  float* hl = hbuf + wslot * (16 * 132);

  int arow = ebase + m;
  if (arow > nedges - 1) arow = nedges - 1;
  const float* A = eattr + (size_t)arow * NGC;

  v8f acc[8];
  zero_acc(acc);

  // GEMM1: [16 x 50] @ [50 x 128] with K padded 50 -> 52
  for (int kk = 0; kk < 48; kk += 4) {
    v2f a = *(const v2f*)(A + kk + koff);
#pragma unroll
    for (int jt = 0; jt < 8; ++jt) {
      v2f b = *(const v2f*)(w1 + (size_t)(jt * 16 + m) * NGC + kk + koff);
      acc[jt] = wmma4(a, b, acc[jt]);
    }
  }
  {  // tail: k=48,49 live in lanes 0-15; lanes 16-31 (k=50,51) contribute zero
    v2f a = {0.0f, 0.0f};
    if (half == 0) a = *(const v2f*)(A + 48);
#pragma unroll
    for (int jt = 0; jt < 8; ++jt) {
      v2f b = {0.0f, 0.0f};
      if (half == 0) b = *(const v2f*)(w1 + (size_t)(jt * 16 + m) * NGC + 48);
      acc[jt] = wmma4(a, b, acc[jt]);
    }
  }

  // bias + shifted-softplus, stage h into LDS (re-layout C-frag -> A-frag)
#pragma unroll
  for (int jt = 0; jt < 8; ++jt) {
    int n = jt * 16 + m;
    float bias = b1[n];
#pragma unroll
    for (int g = 0; g < 8; ++g)
      hl[(g + 8 * half) * 132 + n] = ssp(acc[jt][g] + bias);
  }

  // GEMM2: W = h @ w2.T  (K=128), A fragments from LDS
  zero_acc(acc);
  for (int kk = 0; kk < HIDN; kk += 4) {
    v2f a = *(const v2f*)(hl + m * 132 + kk + koff);
#pragma unroll
    for (int jt = 0; jt < 8; ++jt) {
      v2f b = *(const v2f*)(w2 + (size_t)(jt * 16 + m) * HIDN + kk + koff);
      acc[jt] = wmma4(a, b, acc[jt]);
    }
  }

  // gather xf rows (L2-resident, 25.6 MB) and scatter via hw fp32 atomics
  const int* rowi = ei;
  const int* coli = ei + nedges;
#pragma unroll
  for (int g = 0; g < 8; ++g) {
    int e = ebase + g + 8 * half;
    if (e < nedges) {
      int r = rowi[e];
      int c = coli[e];
      const float* xr = xf + (size_t)r * HIDN;
      const float* xc = xf + (size_t)c * HIDN;
      float* ar = aggr + (size_t)r * HIDN;
      float* ac = aggr + (size_t)c * HIDN;
#pragma unroll
      for (int jt = 0; jt < 8; ++jt) {
        int n = jt * 16 + m;
        float wv = acc[jt][g] + b2[n];
        atomic_add_f32(ac + n, wv * xr[n]);  // msg1 = W*xf[row] -> col
        atomic_add_f32(ar + n, wv * xc[n]);  // msg2 = W*xf[col] -> row
      }
    }
  }
}

// ---------------------------------------------------------------------------
// out = ssp(aggr @ lin2_w.T + lin2_b) @ lin_w.T + lin_b   (fused, K=128 twice)
__global__ __launch_bounds__(128) void node_out_kernel(
    const float* __restrict__ aggr, const float* __restrict__ w2,
    const float* __restrict__ b2, const float* __restrict__ w3,
    const float* __restrict__ b3, float* __restrict__ out, int nrows) {
  __shared__ float hbuf[4 * 16 * 132];

  const int lane = threadIdx.x & 31;
  const int m = lane & 15;
  const int half = lane >> 4;
  const int koff = half * 2;
  const int wslot = threadIdx.x >> 5;
  const int wid = blockIdx.x * 4 + wslot;
  const int rbase = wid * 16;
  if (rbase >= nrows) return;
  float* hl = hbuf + wslot * (16 * 132);

  int arow = rbase + m;
  if (arow > nrows - 1) arow = nrows - 1;
  const float* A = aggr + (size_t)arow * HIDN;

  v8f acc[8];
  zero_acc(acc);

  for (int kk = 0; kk < HIDN; kk += 4) {
    v2f a = *(const v2f*)(A + kk + koff);
#pragma unroll
    for (int jt = 0; jt < 8; ++jt) {
      v2f b = *(const v2f*)(w2 + (size_t)(jt * 16 + m) * HIDN + kk + koff);
      acc[jt] = wmma4(a, b, acc[jt]);
    }
  }

#pragma unroll
  for (int jt = 0; jt < 8; ++jt) {
    int n = jt * 16 + m;
    float bias = b2[n];
#pragma unroll
    for (int g = 0; g < 8; ++g)
      hl[(g + 8 * half) * 132 + n] = ssp(acc[jt][g] + bias);
  }

  zero_acc(acc);
  for (int kk = 0; kk < HIDN; kk += 4) {
    v2f a = *(const v2f*)(hl + m * 132 + kk + koff);
#pragma unroll
    for (int jt = 0; jt < 8; ++jt) {
      v2f b = *(const v2f*)(w3 + (size_t)(jt * 16 + m) * HIDN + kk + koff);
      acc[jt] = wmma4(a, b, acc[jt]);
    }
  }

#pragma unroll
  for (int g = 0; g < 8; ++g) {
    int row = rbase + g + 8 * half;
    if (row < nrows) {
#pragma unroll
      for (int jt = 0; jt < 8; ++jt) {
        int n = jt * 16 + m;
        out[(size_t)row * HIDN + n] = acc[jt][g] + b3[n];
      }
    }
  }
}

// ---------------------------------------------------------------------------
extern "C" void kernel_launch(void* const* d_in, const int* in_sizes, int n_in,
                              void* d_out, int out_size, void* d_ws,
                              size_t ws_size, hipStream_t stream) {
  const float* x      = (const float*)d_in[0];
  const int*   ei     = (const int*)d_in[1];
  // d_in[2] = edge_weight (unused by the reference computation)
  const float* eattr  = (const float*)d_in[3];
  const float* lin1_w = (const float*)d_in[4];
  const float* mlp_w1 = (const float*)d_in[5];
  const float* mlp_b1 = (const float*)d_in[6];
  const float* mlp_w2 = (const float*)d_in[7];
  const float* mlp_b2 = (const float*)d_in[8];
  const float* lin2_w = (const float*)d_in[9];
  const float* lin2_b = (const float*)d_in[10];
  const float* lin_w  = (const float*)d_in[11];
  const float* lin_b  = (const float*)d_in[12];
  float* out = (float*)d_out;

  const int N = in_sizes[0] / HIDN;  // 50000
  const int E = in_sizes[1] / 2;     // 800000

  float* xf   = (float*)d_ws;              // [N,128]  (25.6 MB)
  float* aggr = xf + (size_t)N * HIDN;     // [N,128]  (25.6 MB)

  zero_kernel<<<1024, 256, 0, stream>>>(aggr, (long)N * HIDN);

  const int nodeWaves = (N + 15) / 16;
  xf_kernel<<<(nodeWaves + 7) / 8, 256, 0, stream>>>(x, lin1_w, xf, N);

  const int edgeWaves = (E + 15) / 16;
  edge_kernel<<<(edgeWaves + 3) / 4, 128, 0, stream>>>(
      eattr, ei, mlp_w1, mlp_b1, mlp_w2, mlp_b2, xf, aggr, E);

  node_out_kernel<<<(nodeWaves + 3) / 4, 128, 0, stream>>>(
      aggr, lin2_w, lin2_b, lin_w, lin_b, out, N);
}